// ROCModel_21251498180788
// MI455X (gfx1250) — compile-verified
//
#include <hip/hip_runtime.h>

// Problem constants (match reference): B=4096, N=16384, M=100
constexpr int M_PTS  = 100;   // linspace points
constexpr int NBINS  = 101;   // M+1 histogram buckets (bucket 0 is dropped)
constexpr int TPB    = 256;   // histogram block size (8 wave32)
constexpr int NSUB   = TPB/2; // one private histogram per pair of threads

typedef float v2f __attribute__((ext_vector_type(2)));
typedef float v4f __attribute__((ext_vector_type(4)));
typedef float v8f __attribute__((ext_vector_type(8)));

// ---------------------------------------------------------------------------
// Kernel 1: weighted histogram per row.
// One workgroup per row. Each pair of threads owns a private 101-bin LDS
// histogram (128*101*4 = 51,712 B). Stride 101 (odd) => 32 lanes hit 32
// distinct banks; ds_add_f32 atomics within a pair almost never collide.
// Streaming loads are non-temporal B128 (x is read once; keep L2 for hist).
// ---------------------------------------------------------------------------
__global__ __launch_bounds__(TPB) void hist_kernel(const float* __restrict__ x,
                                                   float* __restrict__ hist,
                                                   int ncol)
{
    __shared__ float lh[NSUB * NBINS];

    const int tid = threadIdx.x;
    const int row = blockIdx.x;

    for (int i = tid; i < NSUB * NBINS; i += TPB) lh[i] = 0.0f;
    __syncthreads();

    const v4f* __restrict__ xr = (const v4f*)(x + (size_t)row * ncol);
    float* myh = lh + (tid >> 1) * NBINS;

    const int nv4 = ncol >> 10;           // float4 elements per thread (=16)
    for (int i = 0; i < nv4; i += 4) {
        // 4 independent B128 loads in flight per wave for MLP
        v4f q0 = __builtin_nontemporal_load(&xr[tid + TPB * (i + 0)]);
        v4f q1 = __builtin_nontemporal_load(&xr[tid + TPB * (i + 1)]);
        v4f q2 = __builtin_nontemporal_load(&xr[tid + TPB * (i + 2)]);
        v4f q3 = __builtin_nontemporal_load(&xr[tid + TPB * (i + 3)]);
        float vals[16] = {q0.x,q0.y,q0.z,q0.w, q1.x,q1.y,q1.z,q1.w,
                          q2.x,q2.y,q2.z,q2.w, q3.x,q3.y,q3.z,q3.w};
        #pragma unroll
        for (int c = 0; c < 16; ++c) {
            float f = vals[c];
            // bucket = #{t_i < f}, t_i = i/99  =>  ceil(99*f), clamped
            int j = (int)ceilf(f * (float)(M_PTS - 1));
            j = j < 0 ? 0 : (j > M_PTS ? M_PTS : j);
            atomicAdd(&myh[j], f);        // ds_add_f32, no return
        }
    }
    __syncthreads();

    // Reduce 128 private histograms -> totals; threads 0..100, 4-way unrolled
    if (tid < NBINS) {
        float s0 = 0.f, s1 = 0.f, s2 = 0.f, s3 = 0.f;
        #pragma unroll 4
        for (int h = 0; h < NSUB; h += 4) {
            s0 += lh[(h + 0) * NBINS + tid];
            s1 += lh[(h + 1) * NBINS + tid];
            s2 += lh[(h + 2) * NBINS + tid];
            s3 += lh[(h + 3) * NBINS + tid];
        }
        hist[(size_t)row * NBINS + tid] = (s0 + s1) + (s2 + s3);
    }
}

// ---------------------------------------------------------------------------
// Kernel 2: per-16-row postprocessing, one wave32 per block.
//   full[r][k] = hist[r][k+1], k=0..99 (full[99] = tail = hist[100])
//   roc = full x S, S[k][j] = (k>=j)  -> V_WMMA_F32_16X16X4_F32 accumulation
//   roc_deriv, trapz values in closed form from full.
// ---------------------------------------------------------------------------
constexpr int FSTR = 113;   // odd LDS row stride (bank-conflict free)

__global__ __launch_bounds__(32) void post_kernel(const float* __restrict__ hist,
                                                  const float* __restrict__ y,
                                                  float* __restrict__ out,
                                                  int btot)
{
    __shared__ float fullm[16 * FSTR];
    __shared__ float ytile[16];

    const int lane = threadIdx.x;
    const int row0 = blockIdx.x * 16;

    // Load 16x101 hist tile; c=0 is dropped; c in 1..100 -> k=c-1 in 0..99
    for (int i = lane; i < 16 * NBINS; i += 32) {
        int r = i / NBINS, c = i % NBINS;
        float v = hist[(size_t)(row0 + r) * NBINS + c];
        if (c >= 1) fullm[r * FSTR + (c - 1)] = v;
    }
    if (lane < 16) ytile[lane] = y[row0 + lane];
    __syncthreads();

    const int hi = lane >> 4;     // half-wave id
    const int lo = lane & 15;

    float invy[8];
    #pragma unroll
    for (int v = 0; v < 8; ++v) invy[v] = 1.0f / ytile[v + 8 * hi];

    float* out_roc  = out;
    float* out_rocd = out + (size_t)btot * M_PTS;
    float* out_t1   = out + (size_t)btot * (2 * M_PTS);
    float* out_t2   = out_t1 + btot;

    // roc = full(16x100) x S(100x100), tiled: N in 7 tiles of 16, K in 4s.
    // All-zero B chunks (4kc+3 < 16n) skipped -> kc starts at 4n.
    for (int n = 0; n < 7; ++n) {
        v8f acc = {};
        const int jg = 16 * n + lo;                  // global output column
        for (int kc = 4 * n; kc < 25; ++kc) {
            const int kb = 4 * kc + 2 * hi;          // A: lanes0-15 K0,K1; lanes16-31 K2,K3
            v2f a;
            a.x = fullm[lo * FSTR + kb];
            a.y = fullm[lo * FSTR + kb + 1];
            v2f b;                                   // B (4x16): V0->K(2hi), V1->K(2hi+1)
            b.x = (kb     >= jg) ? 1.0f : 0.0f;
            b.y = (kb + 1 >= jg) ? 1.0f : 0.0f;
            acc = __builtin_amdgcn_wmma_f32_16x16x4_f32(
                      false, a, false, b, (short)0, acc, false, false);
        }
        if (jg < M_PTS) {
            #pragma unroll
            for (int v = 0; v < 8; ++v) {            // C/D: VGPR v -> row v+8*hi
                int m = v + 8 * hi;
                out_roc[(size_t)(row0 + m) * M_PTS + jg] = acc[v] * invy[v];
            }
        }
    }

    // roc_deriv + both trapz integrals (closed form), one row per lane 0..15
    if (lane < 16) {
        const float* f = fullm + lane * FSTR;
        const float iy = 1.0f / ytile[lane];
        float* rd = out_rocd + (size_t)(row0 + lane) * M_PTS;
        float s1 = 0.0f, s2 = 0.0f;
        #pragma unroll 4
        for (int k = 0; k < M_PTS - 1; ++k) {        // k = 0..98
            float fv = f[k];
            s1 += ((float)k + 0.5f) * fv;
            s2 += fv;
            rd[k] = fv * iy;
        }
        const float f99 = f[99], f98 = f[98], f97 = f[97], f0 = f[0];
        s1 += 99.5f * f99;                           // include k=99 term
        rd[99] = 0.5f * (f97 + f98) * iy;
        // trapz(g) = sum(g) - 0.5*g[0] - 0.5*g[99]
        out_t1[row0 + lane] = (s1 - 0.5f * f99) * iy;
        out_t2[row0 + lane] = (s2 - 0.5f * f0 + 0.25f * (f97 + f98)) * iy;
    }
}

// ---------------------------------------------------------------------------
extern "C" void kernel_launch(void* const* d_in, const int* in_sizes, int n_in,
                              void* d_out, int out_size, void* d_ws, size_t ws_size,
                              hipStream_t stream)
{
    const float* x = (const float*)d_in[0];
    const float* y = (const float*)d_in[1];
    float* out = (float*)d_out;

    const int btot = in_sizes[1];            // 4096 rows
    const int ncol = in_sizes[0] / btot;     // 16384 cols

    float* hist = (float*)d_ws;              // btot * 101 f32 (~1.65 MB)

    hist_kernel<<<btot, TPB, 0, stream>>>(x, hist, ncol);
    post_kernel<<<btot / 16, 32, 0, stream>>>(hist, y, out, btot);
}